// ScaleDotProductAtt_71176198029372
// MI455X (gfx1250) — compile-verified
//
#include <hip/hip_runtime.h>
#include <cstdint>

// Problem constants (reference: B=2, S=2048, H=16, E=1024, D=64)
#define B_  2
#define S_  2048
#define H_  16
#define E_  1024
#define D_  64
#define BH_ (B_ * H_)     // 32
#define NT_ (S_ / 16)     // 128 16-wide tiles along S
#define SCH_ 16           // s-chunks for the v-sum partial reduction

typedef __attribute__((ext_vector_type(2))) float v2f;
typedef __attribute__((ext_vector_type(8))) float v8f;

// D = A(16x4 f32) * B(4x16 f32) + C(16x16 f32), exact fp32 on the matrix core.
__device__ __forceinline__ v8f wmma_f32_16x16x4(v2f a, v2f b, v8f c) {
  return __builtin_amdgcn_wmma_f32_16x16x4_f32(false, a, false, b, (short)0, c,
                                               false, false);
}

// Load one fragment set (16 k-steps of K=4) for a 16-row tile of a row-major
// [S, E] matrix. Lane layout: lane.m selects the row, half/vgpr select
// K_local = v + 2*hf.
__device__ __forceinline__ void load_frag16(v2f (&f)[16],
                                            const float* __restrict__ base) {
#pragma unroll
  for (int ks = 0; ks < 16; ++ks)
    f[ks] = *(const v2f*)(base + ks * 4);
}

// ---------------------------------------------------------------------------
// Pass 1: per-row softmax stats via lane-local online softmax.
// One wave owns a 16-query strip and sweeps all 2048 keys with WMMA.
// Q fragments are pre-scaled by 1/32 (exact power-of-two), so the hot loop
// has no score post-scaling. Cross-lane merge happens once at the end.
// Output: stats[row] = (rowmax, 1/rowsum) as float2.
// ---------------------------------------------------------------------------
__global__ __launch_bounds__(256) void qk_stats_kernel(
    const float* __restrict__ q, const float* __restrict__ k,
    v2f* __restrict__ stats) {
  const int wave = threadIdx.x >> 5;
  const int lane = threadIdx.x & 31;
  const int m  = lane & 15;   // A: row M / B: col N
  const int hf = lane >> 4;   // lane half
  const int bh    = blockIdx.x >> 4;
  const int strip = ((blockIdx.x & 15) << 3) + wave;   // 0..127
  const int b = bh >> 4, h = bh & 15;
  const int qRow0 = strip * 16;
  const float scale = 1.0f / 32.0f;  // 1/sqrt(E), exact

  const float* qbase =
      q + ((size_t)b * S_ + (qRow0 + m)) * E_ + h * D_ + 2 * hf;
  const float* kbase0 = k + ((size_t)b * S_ + m) * E_ + h * D_ + 2 * hf;
  const size_t tileStride = (size_t)16 * E_;   // 16 key rows

  v2f aQ[16];
  load_frag16(aQ, qbase);
#pragma unroll
  for (int ks = 0; ks < 16; ++ks) aQ[ks] = aQ[ks] * scale;  // fold the scale

  float lmax[8], lsum[8];
#pragma unroll
  for (int v = 0; v < 8; ++v) { lmax[v] = -3.402823466e38f; lsum[v] = 0.0f; }

  v2f cur[16], nxt[16];
  load_frag16(cur, kbase0);

  for (int kt = 0; kt < NT_; kt += 2) {
    load_frag16(nxt, kbase0 + (size_t)(kt + 1) * tileStride);
    {
      v8f c = {};
#pragma unroll
      for (int ks = 0; ks < 16; ++ks) c = wmma_f32_16x16x4(aQ[ks], cur[ks], c);
#pragma unroll
      for (int v = 0; v < 8; ++v) {
        float s = c[v];
        float nm = fmaxf(lmax[v], s);
        lsum[v] = lsum[v] * __expf(lmax[v] - nm) + __expf(s - nm);
        lmax[v] = nm;
      }
    }
    if (kt + 2 < NT_)
      load_frag16(cur, kbase0 + (size_t)(kt + 2) * tileStride);
    {
      v8f c = {};
#pragma unroll
      for (int ks = 0; ks < 16; ++ks) c = wmma_f32_16x16x4(aQ[ks], nxt[ks], c);
#pragma unroll
      for (int v = 0; v < 8; ++v) {
        float s = c[v];
        float nm = fmaxf(lmax[v], s);
        lsum[v] = lsum[v] * __expf(lmax[v] - nm) + __expf(s - nm);
        lmax[v] = nm;
      }
    }
  }

  // Merge the 16 per-lane partial (max, sum) pairs within each lane half.
#pragma unroll
  for (int v = 0; v < 8; ++v) {
    float mx = lmax[v], sm = lsum[v];
#pragma unroll
    for (int msk = 1; msk <= 8; msk <<= 1) {
      float om = __shfl_xor(mx, msk, 32);
      float os = __shfl_xor(sm, msk, 32);
      float nm = fmaxf(mx, om);
      sm = sm * __expf(mx - nm) + os * __expf(om - nm);
      mx = nm;
    }
    lmax[v] = mx; lsum[v] = sm;
  }
  if (m == 0) {
#pragma unroll
    for (int v = 0; v < 8; ++v) {
      int row = qRow0 + v + 8 * hf;
      v2f st; st.x = lmax[v]; st.y = 1.0f / lsum[v];
      stats[(size_t)bh * S_ + row] = st;
    }
  }
}

// ---------------------------------------------------------------------------
// Pass 2: recompute scores (Q/K are L2-resident), stream normalized attn to
// HBM with non-temporal stores, and fuse column sums in registers — each wave
// owns one 16-key column strip over all query rows, so no atomics needed.
// K fragments pre-scaled by 1/32; (rowmax, 1/rowsum) read as one b64 load.
// ---------------------------------------------------------------------------
__global__ __launch_bounds__(256) void attn_colsum_kernel(
    const float* __restrict__ q, const float* __restrict__ k,
    const v2f* __restrict__ stats, float* __restrict__ attn,
    float* __restrict__ colsum) {
  const int wave = threadIdx.x >> 5;
  const int lane = threadIdx.x & 31;
  const int m  = lane & 15;
  const int hf = lane >> 4;
  const int bh    = blockIdx.x >> 4;
  const int ktile = ((blockIdx.x & 15) << 3) + wave;   // 0..127
  const int b = bh >> 4, h = bh & 15;
  const int key0 = ktile * 16;
  const float scale = 1.0f / 32.0f;

  const float* kbase =
      k + ((size_t)b * S_ + (key0 + m)) * E_ + h * D_ + 2 * hf;
  v2f bK[16];
  load_frag16(bK, kbase);
#pragma unroll
  for (int ks = 0; ks < 16; ++ks) bK[ks] = bK[ks] * scale;  // fold the scale

  const float* qbase0 = q + ((size_t)b * S_ + m) * E_ + h * D_ + 2 * hf;
  const size_t tileStride = (size_t)16 * E_;

  float* attnBH = attn + (size_t)bh * S_ * S_;
  const v2f* statsBH = stats + (size_t)bh * S_;
  float csum = 0.0f;

  v2f cur[16], nxt[16];
  load_frag16(cur, qbase0);

  for (int qt = 0; qt < NT_; qt += 2) {
    load_frag16(nxt, qbase0 + (size_t)(qt + 1) * tileStride);
    {
      v8f c = {};
#pragma unroll
      for (int ks = 0; ks < 16; ++ks) c = wmma_f32_16x16x4(cur[ks], bK[ks], c);
      const int qRow0 = qt * 16;
#pragma unroll
      for (int v = 0; v < 8; ++v) {
        int row = qRow0 + v + 8 * hf;
        v2f st = statsBH[row];
        float p = __expf(c[v] - st.x) * st.y;
        __builtin_nontemporal_store(p, &attnBH[(size_t)row * S_ + key0 + m]);
        csum += p;
      }
    }
    if (qt + 2 < NT_)
      load_frag16(cur, qbase0 + (size_t)(qt + 2) * tileStride);
    {
      v8f c = {};
#pragma unroll
      for (int ks = 0; ks < 16; ++ks) c = wmma_f32_16x16x4(nxt[ks], bK[ks], c);
      const int qRow0 = (qt + 1) * 16;
#pragma unroll
      for (int v = 0; v < 8; ++v) {
        int row = qRow0 + v + 8 * hf;
        v2f st = statsBH[row];
        float p = __expf(c[v] - st.x) * st.y;
        __builtin_nontemporal_store(p, &attnBH[(size_t)row * S_ + key0 + m]);
        csum += p;
      }
    }
  }
  // Lane m and lane m+16 hold the same column (different row halves).
  csum += __shfl_xor(csum, 16, 32);
  if (lane < 16)
    colsum[(size_t)bh * S_ + key0 + lane] = csum;
}

// ---------------------------------------------------------------------------
// v_sum stage 1: partial sums over s-chunks (128 blocks, coalesced reads).
// ---------------------------------------------------------------------------
__global__ __launch_bounds__(256) void vsum_partial_kernel(
    const float* __restrict__ v, float* __restrict__ partial) {
  const int nCols = B_ * E_;                       // 2048
  const int chunk = blockIdx.x % SCH_;
  const int col0  = (blockIdx.x / SCH_) * 256;     // column group
  const int idx   = col0 + threadIdx.x;            // 0..2047
  const int b = idx / E_;
  const int e = idx % E_;
  const int rowsPerChunk = S_ / SCH_;              // 128
  const float* p =
      v + (size_t)b * S_ * E_ + (size_t)(chunk * rowsPerChunk) * E_ + e;
  float acc = 0.0f;
  for (int s = 0; s < rowsPerChunk; ++s) acc += p[(size_t)s * E_];
  partial[(size_t)chunk * nCols + idx] = acc;
}

// v_sum stage 2: reduce the SCH_ partials.
__global__ void vsum_reduce_kernel(const float* __restrict__ partial,
                                   float* __restrict__ vsum) {
  const int nCols = B_ * E_;
  int idx = blockIdx.x * blockDim.x + threadIdx.x;
  if (idx >= nCols) return;
  float acc = 0.0f;
#pragma unroll
  for (int c = 0; c < SCH_; ++c) acc += partial[(size_t)c * nCols + idx];
  vsum[idx] = acc;
}

// ---------------------------------------------------------------------------
// result[b, s, e] = colsum[b, h(e), s] * vsum[b, e]   (outer product, 16 MB)
// ---------------------------------------------------------------------------
__global__ void result_kernel(const float* __restrict__ colsum,
                              const float* __restrict__ vsum,
                              float* __restrict__ out) {
  size_t i = (size_t)blockIdx.x * blockDim.x + threadIdx.x;
  if (i >= (size_t)B_ * S_ * E_) return;
  int e = (int)(i % E_);
  int s = (int)((i / E_) % S_);
  int b = (int)(i / ((size_t)E_ * S_));
  int h = e / D_;
  out[i] = colsum[((size_t)(b * H_ + h)) * S_ + s] * vsum[b * E_ + e];
}

extern "C" void kernel_launch(void* const* d_in, const int* in_sizes, int n_in,
                              void* d_out, int out_size, void* d_ws,
                              size_t ws_size, hipStream_t stream) {
  (void)in_sizes; (void)n_in; (void)out_size; (void)ws_size;
  const float* q = (const float*)d_in[0];
  const float* k = (const float*)d_in[1];
  const float* v = (const float*)d_in[2];

  float* out    = (float*)d_out;
  float* result = out;                                  // [B,S,E]
  float* attn   = out + (size_t)B_ * S_ * E_;           // [B,H,S,S]

  float* ws      = (float*)d_ws;
  v2f*   stats   = (v2f*)ws;                            // BH_*S_ float2
  float* colsum  = ws + (size_t)2 * BH_ * S_;           // BH_*S_ floats
  float* vsum    = colsum + (size_t)BH_ * S_;           // B_*E_ floats
  float* partial = vsum + (size_t)B_ * E_;              // SCH_*B_*E_ floats

  qk_stats_kernel<<<BH_ * 16, 256, 0, stream>>>(q, k, stats);
  attn_colsum_kernel<<<BH_ * 16, 256, 0, stream>>>(q, k, stats, attn, colsum);
  vsum_partial_kernel<<<(B_ * E_ / 256) * SCH_, 256, 0, stream>>>(v, partial);
  vsum_reduce_kernel<<<(B_ * E_ + 255) / 256, 256, 0, stream>>>(partial, vsum);
  result_kernel<<<(int)(((size_t)B_ * S_ * E_ + 255) / 256), 256, 0, stream>>>(
      colsum, vsum, result);
}